// DualDilatedEdgeGraphConvBlock_23089744183609
// MI455X (gfx1250) — compile-verified
//
#include <hip/hip_runtime.h>

// Problem constants (match reference)
#define Bb   4
#define Nn   4096
#define BN   (Bb * Nn)      // 16384
#define Cc   16
#define C2   32
#define C4   64
#define HIDd 64
#define OUTd 64
#define ALLK 128
#define Kk   32
#define KD   8
#define KDP  6
#define KE   14             // KD + KDP
#define NEG  0.2f
#define FLTMAX 3.402823466e+38f

typedef __attribute__((ext_vector_type(2))) float v2f;
typedef __attribute__((ext_vector_type(8))) float v8f;

__device__ __forceinline__ v8f v8zero() {
  v8f z = {0.f,0.f,0.f,0.f,0.f,0.f,0.f,0.f};
  return z;
}

// D = A(16x4 f32) * B(4x16 f32) + C, CDNA5 V_WMMA_F32_16X16X4_F32.
// A layout: lanes 0-15 -> row M=lane, {K0,K1}; lanes 16-31 -> row M=lane-16, {K2,K3}
// B layout: lanes 0-15 -> col N=lane, {K0,K1}; lanes 16-31 -> col N=lane-16, {K2,K3}
// D layout: VGPR i: lanes 0-15 -> M=i, lanes 16-31 -> M=i+8; N = lane%16
__device__ __forceinline__ v8f wmma4(v2f a, v2f b, v8f c) {
#if __has_builtin(__builtin_amdgcn_wmma_f32_16x16x4_f32)
  return __builtin_amdgcn_wmma_f32_16x16x4_f32(false, a, false, b, (short)0, c,
                                               false, false);
#else
  // Shuffle-emulated fallback with identical lane layout (host pass / safety).
  int l = threadIdx.x & 31;
  int Mbase = (l < 16) ? 0 : 8;
  int Ncol = l & 15;
  float b0 = __shfl(b.x, Ncol, 32),      b1 = __shfl(b.y, Ncol, 32);
  float b2 = __shfl(b.x, Ncol + 16, 32), b3 = __shfl(b.y, Ncol + 16, 32);
  #pragma unroll
  for (int i = 0; i < 8; ++i) {
    int M = Mbase + i;
    float a0 = __shfl(a.x, M, 32),      a1 = __shfl(a.y, M, 32);
    float a2 = __shfl(a.x, M + 16, 32), a3 = __shfl(a.y, M + 16, 32);
    c[i] += a0 * b0 + a1 * b1 + a2 * b2 + a3 * b3;
  }
  return c;
#endif
}

// ---------------------------------------------------------------------------
// K1: pos pairwise distances (16 rows x 4096 cols in 256KB LDS, WMMA f32
// 16x16x4 Gram tiles), then wave-per-row top-128 extraction.
// Writes idx_l (ascending distance, self first) and idx_pos slots of idx1.
// ---------------------------------------------------------------------------
__global__ void k_dist_pos(const float* __restrict__ pos,
                           int* __restrict__ idx_l,
                           int* __restrict__ idx1,
                           float* __restrict__ out_idx) {
  extern __shared__ float lds[];                 // 16 * 4096 floats = 256 KB
  const int b  = blockIdx.x >> 8;                // N/16 = 256 row-blocks
  const int r0 = (blockIdx.x & 255) << 4;
  const int wave = threadIdx.x >> 5;
  const int lane = threadIdx.x & 31;
  const int m = lane & 15;
  const bool lo = lane < 16;

  // A operand: rows r0..r0+15 of pos, padded K=4
  const int row = r0 + m;
  const float* pr = pos + (size_t)(b * Nn + row) * 3;
  float p0 = pr[0], p1 = pr[1], p2 = pr[2];
  v2f areg;
  if (lo) { areg.x = p0; areg.y = p1; } else { areg.x = p2; areg.y = 0.f; }
  float sqm = p0 * p0 + p1 * p1 + p2 * p2;
  float sA[8];
  #pragma unroll
  for (int i = 0; i < 8; ++i) sA[i] = __shfl(sqm, lo ? i : (8 + i), 32);

  for (int cb = wave; cb < 256; cb += 16) {      // 16 waves cover 256 col-tiles
    const int col = (cb << 4) + m;
    const float* pc = pos + (size_t)(b * Nn + col) * 3;
    float q0 = pc[0], q1 = pc[1], q2 = pc[2];
    v2f breg;
    if (lo) { breg.x = q0; breg.y = q1; } else { breg.x = q2; breg.y = 0.f; }
    float sqj = q0 * q0 + q1 * q1 + q2 * q2;
    v8f acc = v8zero();
    acc = wmma4(areg, breg, acc);
    #pragma unroll
    for (int i = 0; i < 8; ++i) {
      int M = lo ? i : (8 + i);
      float d = fmaxf(sA[i] + sqj - 2.f * acc[i], 0.f);
      lds[M * Nn + (cb << 4) + m] = d;
    }
  }
  __syncthreads();

  // Selection: wave w extracts 128 ascending minima of row w.
  volatile float* drow = lds + wave * Nn;
  const int rbase = b * Nn + r0 + wave;
  for (int j = 0; j < ALLK; ++j) {
    float lv = FLTMAX; int li = 0;
    for (int t = lane; t < Nn; t += 32) {
      float v = drow[t];
      if (v < lv) { lv = v; li = t; }
    }
    #pragma unroll
    for (int off = 16; off >= 1; off >>= 1) {
      float ov = __shfl_xor(lv, off, 32);
      int   oi = __shfl_xor(li, off, 32);
      if (ov < lv || (ov == lv && oi < li)) { lv = ov; li = oi; }
    }
    if (lane == 0) {
      idx_l[(size_t)rbase * ALLK + j] = li;
      if (j >= 1 && j <= KDP) {                  // pos-kNN excl self
        idx1[rbase * KE + KD + (j - 1)] = li;
        out_idx[(size_t)rbase * KE + KD + (j - 1)] = (float)li;
      }
      drow[li] = FLTMAX;
    }
  }
}

// ---------------------------------------------------------------------------
// K2: feature pairwise distances (C=16 -> 4 WMMA k-steps), top-9, drop self.
// ---------------------------------------------------------------------------
__global__ void k_dist_feat(const float* __restrict__ x,
                            int* __restrict__ idx1,
                            float* __restrict__ out_idx) {
  extern __shared__ float lds[];
  const int b  = blockIdx.x >> 8;
  const int r0 = (blockIdx.x & 255) << 4;
  const int wave = threadIdx.x >> 5;
  const int lane = threadIdx.x & 31;
  const int m = lane & 15;
  const bool lo = lane < 16;

  const int row = r0 + m;
  const float* xr = x + (size_t)(b * Nn + row) * Cc;
  float sqm = 0.f;
  #pragma unroll
  for (int c = 0; c < Cc; ++c) sqm += xr[c] * xr[c];
  v2f aK[4];
  #pragma unroll
  for (int kb = 0; kb < 4; ++kb) {
    int c0 = 4 * kb + (lo ? 0 : 2);
    aK[kb].x = xr[c0]; aK[kb].y = xr[c0 + 1];
  }
  float sA[8];
  #pragma unroll
  for (int i = 0; i < 8; ++i) sA[i] = __shfl(sqm, lo ? i : (8 + i), 32);

  for (int cb = wave; cb < 256; cb += 16) {
    const int col = (cb << 4) + m;
    const float* xc = x + (size_t)(b * Nn + col) * Cc;
    float sqj = 0.f;
    #pragma unroll
    for (int c = 0; c < Cc; ++c) sqj += xc[c] * xc[c];
    v8f acc = v8zero();
    #pragma unroll
    for (int kb = 0; kb < 4; ++kb) {
      int c0 = 4 * kb + (lo ? 0 : 2);
      v2f bK; bK.x = xc[c0]; bK.y = xc[c0 + 1];
      acc = wmma4(aK[kb], bK, acc);
    }
    #pragma unroll
    for (int i = 0; i < 8; ++i) {
      int M = lo ? i : (8 + i);
      float d = fmaxf(sA[i] + sqj - 2.f * acc[i], 0.f);
      lds[M * Nn + (cb << 4) + m] = d;
    }
  }
  __syncthreads();

  volatile float* drow = lds + wave * Nn;
  const int rbase = b * Nn + r0 + wave;
  for (int j = 0; j <= KD; ++j) {                // 9 minima, drop self (j=0)
    float lv = FLTMAX; int li = 0;
    for (int t = lane; t < Nn; t += 32) {
      float v = drow[t];
      if (v < lv) { lv = v; li = t; }
    }
    #pragma unroll
    for (int off = 16; off >= 1; off >>= 1) {
      float ov = __shfl_xor(lv, off, 32);
      int   oi = __shfl_xor(li, off, 32);
      if (ov < lv || (ov == lv && oi < li)) { lv = ov; li = oi; }
    }
    if (lane == 0) {
      if (j >= 1) {
        idx1[rbase * KE + (j - 1)] = li;
        out_idx[(size_t)rbase * KE + (j - 1)] = (float)li;
      }
      drow[li] = FLTMAX;
    }
  }
}

// ---------------------------------------------------------------------------
// K3: farthest point sampling, one wave per point group (128 candidates -> 32)
// ---------------------------------------------------------------------------
__global__ void k_fps(const float* __restrict__ pos,
                      const int* __restrict__ idx_l,
                      int* __restrict__ idx_fps) {
  const int wave = threadIdx.x >> 5;
  const int lane = threadIdx.x & 31;
  const int g = blockIdx.x * 8 + wave;
  const int b = g / Nn;
  const int* il = idx_l + (size_t)g * ALLK;

  float px[4], py[4], pz[4], dmin[4];
  #pragma unroll
  for (int s = 0; s < 4; ++s) {
    int idx = il[lane * 4 + s];
    const float* p = pos + (size_t)(b * Nn + idx) * 3;
    px[s] = p[0]; py[s] = p[1]; pz[s] = p[2];
    dmin[s] = 1e10f;
  }
  int far = 0;
  for (int i = 0; i < Kk; ++i) {
    if (lane == 0) idx_fps[(size_t)g * Kk + i] = il[far];
    const int owner = far >> 2, slot = far & 3;   // uniform
    float sx, sy, sz;
    switch (slot) {
      case 0: sx = px[0]; sy = py[0]; sz = pz[0]; break;
      case 1: sx = px[1]; sy = py[1]; sz = pz[1]; break;
      case 2: sx = px[2]; sy = py[2]; sz = pz[2]; break;
      default: sx = px[3]; sy = py[3]; sz = pz[3]; break;
    }
    float cx = __shfl(sx, owner, 32);
    float cy = __shfl(sy, owner, 32);
    float cz = __shfl(sz, owner, 32);
    float bv = -1.f; int bj = 0;
    #pragma unroll
    for (int s = 0; s < 4; ++s) {
      float dx = px[s] - cx, dy = py[s] - cy, dz = pz[s] - cz;
      dmin[s] = fminf(dmin[s], dx * dx + dy * dy + dz * dz);
      if (dmin[s] > bv) { bv = dmin[s]; bj = lane * 4 + s; }
    }
    #pragma unroll
    for (int off = 16; off >= 1; off >>= 1) {
      float ov = __shfl_xor(bv, off, 32);
      int   oj = __shfl_xor(bj, off, 32);
      if (ov > bv || (ov == bv && oj < bj)) { bv = ov; bj = oj; }
    }
    far = bj;
  }
}

// ---------------------------------------------------------------------------
// K4: h1 = e1 @ W1^T, accumulate per-channel sum/sumsq (BN stats pass).
// One 16-row tile per wave; e1 staged in per-wave LDS.
// ---------------------------------------------------------------------------
__global__ void k_h1_stats(const float* __restrict__ x,
                           const int* __restrict__ idx1,
                           const float* __restrict__ W1,
                           float* __restrict__ stats) {  // [0..31]=sum [32..63]=sumsq
  __shared__ float e1s[8][16][C2];
  const int wave = threadIdx.x >> 5;
  const int lane = threadIdx.x & 31;
  const int m = lane & 15;
  const bool lo = lane < 16;
  const int t = blockIdx.x * 8 + wave;

  if (lo) {
    int r = t * 16 + lane;
    int bn = r / KE, k = r % KE;
    int b = bn / Nn;
    int nidx = idx1[bn * KE + k];
    const float* xc = x + (size_t)bn * Cc;
    const float* xn = x + (size_t)(b * Nn + nidx) * Cc;
    #pragma unroll
    for (int c = 0; c < Cc; ++c) {
      float cv = xc[c];
      e1s[wave][lane][c]      = xn[c] - cv;
      e1s[wave][lane][Cc + c] = cv;
    }
  }
  __syncthreads();

  v8f acc[2]; acc[0] = v8zero(); acc[1] = v8zero();
  #pragma unroll
  for (int kb = 0; kb < 8; ++kb) {
    int c0 = 4 * kb + (lo ? 0 : 2);
    v2f a; a.x = e1s[wave][m][c0]; a.y = e1s[wave][m][c0 + 1];
    #pragma unroll
    for (int ct = 0; ct < 2; ++ct) {
      int o = ct * 16 + m;
      v2f bw; bw.x = W1[o * C2 + c0]; bw.y = W1[o * C2 + c0 + 1];
      acc[ct] = wmma4(a, bw, acc[ct]);
    }
  }
  #pragma unroll
  for (int ct = 0; ct < 2; ++ct) {
    float s = 0.f, ss = 0.f;
    #pragma unroll
    for (int i = 0; i < 8; ++i) { float v = acc[ct][i]; s += v; ss += v * v; }
    s  += __shfl_xor(s, 16, 32);
    ss += __shfl_xor(ss, 16, 32);
    if (lo) {
      int ch = ct * 16 + m;
      atomicAdd(&stats[ch], s);
      atomicAdd(&stats[C2 + ch], ss);
    }
  }
}

// Shared tiny kernel: mu/var -> (scale, bias) per channel.
__global__ void k_finalize(const float* __restrict__ sum,
                           const float* __restrict__ sq,
                           const float* __restrict__ gamma,
                           const float* __restrict__ beta,
                           float cntInv, int nch, float* __restrict__ sb) {
  int c = threadIdx.x;
  if (c < nch) {
    float mu  = sum[c] * cntInv;
    float var = sq[c] * cntInv - mu * mu;
    float sc  = gamma[c] * rsqrtf(var + 1e-5f);
    sb[c] = sc;
    sb[nch + c] = beta[c] - mu * sc;
  }
}

// ---------------------------------------------------------------------------
// K6: recompute h1, normalize + lrelu, max over k=14 -> f1. One point per wave
// (rows padded to 16; padded rows excluded from the max).
// ---------------------------------------------------------------------------
__global__ void k_h1_apply(const float* __restrict__ x,
                           const int* __restrict__ idx1,
                           const float* __restrict__ W1,
                           const float* __restrict__ sb1,
                           float* __restrict__ f1) {
  __shared__ float e1s[8][16][C2];
  const int wave = threadIdx.x >> 5;
  const int lane = threadIdx.x & 31;
  const int m = lane & 15;
  const bool lo = lane < 16;
  const int bn = blockIdx.x * 8 + wave;
  const int b = bn / Nn;

  if (lo) {
    int k = lane;
    if (k < KE) {
      int nidx = idx1[bn * KE + k];
      const float* xc = x + (size_t)bn * Cc;
      const float* xn = x + (size_t)(b * Nn + nidx) * Cc;
      #pragma unroll
      for (int c = 0; c < Cc; ++c) {
        float cv = xc[c];
        e1s[wave][lane][c]      = xn[c] - cv;
        e1s[wave][lane][Cc + c] = cv;
      }
    } else {
      #pragma unroll
      for (int c = 0; c < C2; ++c) e1s[wave][lane][c] = 0.f;
    }
  }
  __syncthreads();

  v8f acc[2]; acc[0] = v8zero(); acc[1] = v8zero();
  #pragma unroll
  for (int kb = 0; kb < 8; ++kb) {
    int c0 = 4 * kb + (lo ? 0 : 2);
    v2f a; a.x = e1s[wave][m][c0]; a.y = e1s[wave][m][c0 + 1];
    #pragma unroll
    for (int ct = 0; ct < 2; ++ct) {
      int o = ct * 16 + m;
      v2f bw; bw.x = W1[o * C2 + c0]; bw.y = W1[o * C2 + c0 + 1];
      acc[ct] = wmma4(a, bw, acc[ct]);
    }
  }
  const int imax = lo ? 8 : 6;                    // exclude padded rows 14,15
  #pragma unroll
  for (int ct = 0; ct < 2; ++ct) {
    int ch = ct * 16 + m;
    float sc = sb1[ch], bi = sb1[C2 + ch];
    float mx = -FLTMAX;
    #pragma unroll
    for (int i = 0; i < 8; ++i) {
      if (i < imax) {
        float v = acc[ct][i] * sc + bi;
        v = v > 0.f ? v : NEG * v;
        mx = fmaxf(mx, v);
      }
    }
    mx = fmaxf(mx, __shfl_xor(mx, 16, 32));
    if (lo) f1[(size_t)bn * C2 + ch] = mx;
  }
}

// Common device helper: stage one 16-row e2 tile for point g into buf[wave].
__device__ __forceinline__ void stage_e2(float (*buf)[C4],
                                         const float* __restrict__ f1,
                                         const int* __restrict__ idx_fps,
                                         int g, int b, int krow, int lane) {
  if (lane < 16) {
    int nidx = idx_fps[(size_t)g * Kk + krow];
    const float* fc = f1 + (size_t)g * C2;
    const float* fn = f1 + (size_t)(b * Nn + nidx) * C2;
    #pragma unroll
    for (int c = 0; c < C2; ++c) {
      float cv = fc[c];
      buf[lane][c]      = fn[c] - cv;
      buf[lane][C2 + c] = cv;
    }
  }
}

__device__ __forceinline__ void gemm64(v8f acc[4], const float (*buf)[C4],
                                       const float* __restrict__ W,
                                       int m, bool lo) {
  #pragma unroll
  for (int kb = 0; kb < 16; ++kb) {
    int c0 = 4 * kb + (lo ? 0 : 2);
    v2f a; a.x = buf[m][c0]; a.y = buf[m][c0 + 1];
    #pragma unroll
    for (int ct = 0; ct < 4; ++ct) {
      int o = ct * 16 + m;
      v2f bw; bw.x = W[o * C4 + c0]; bw.y = W[o * C4 + c0 + 1];
      acc[ct] = wmma4(a, bw, acc[ct]);
    }
  }
}

// K7: h2 = e2 @ W2^T stats pass. One 16-row tile per wave.
__global__ void k_h2_stats(const float* __restrict__ f1,
                           const int* __restrict__ idx_fps,
                           const float* __restrict__ W2,
                           float* __restrict__ stats) { // [0..63]=sum [64..127]=sumsq
  __shared__ float e2s[8][16][C4];
  const int wave = threadIdx.x >> 5;
  const int lane = threadIdx.x & 31;
  const int m = lane & 15;
  const bool lo = lane < 16;
  const int t = blockIdx.x * 8 + wave;
  const int r = t * 16 + (lane & 15);
  const int g = (t * 16) >> 5;                    // all 16 rows share g
  const int b = g / Nn;
  stage_e2(e2s[wave], f1, idx_fps, g, b, r & 31, lane);
  __syncthreads();

  v8f acc[4];
  #pragma unroll
  for (int ct = 0; ct < 4; ++ct) acc[ct] = v8zero();
  gemm64(acc, e2s[wave], W2, m, lo);

  #pragma unroll
  for (int ct = 0; ct < 4; ++ct) {
    float s = 0.f, ss = 0.f;
    #pragma unroll
    for (int i = 0; i < 8; ++i) { float v = acc[ct][i]; s += v; ss += v * v; }
    s  += __shfl_xor(s, 16, 32);
    ss += __shfl_xor(ss, 16, 32);
    if (lo) {
      int ch = ct * 16 + m;
      atomicAdd(&stats[ch], s);
      atomicAdd(&stats[C4 + ch], ss);
    }
  }
}

// K9: recompute h2, normalize+lrelu -> h2n in LDS, h3 = h2n @ W3^T stats pass.
__global__ void k_h3_stats(const float* __restrict__ f1,
                           const int* __restrict__ idx_fps,
                           const float* __restrict__ W2,
                           const float* __restrict__ W3,
                           const float* __restrict__ sb2,
                           float* __restrict__ stats) { // sum3/sq3
  __shared__ float buf[8][16][C4];
  const int wave = threadIdx.x >> 5;
  const int lane = threadIdx.x & 31;
  const int m = lane & 15;
  const bool lo = lane < 16;
  const int t = blockIdx.x * 8 + wave;
  const int r = t * 16 + (lane & 15);
  const int g = (t * 16) >> 5;
  const int b = g / Nn;
  stage_e2(buf[wave], f1, idx_fps, g, b, r & 31, lane);
  __syncthreads();

  v8f acc[4];
  #pragma unroll
  for (int ct = 0; ct < 4; ++ct) acc[ct] = v8zero();
  gemm64(acc, buf[wave], W2, m, lo);
  __syncthreads();
  #pragma unroll
  for (int ct = 0; ct < 4; ++ct) {
    int ch = ct * 16 + m;
    float sc = sb2[ch], bi = sb2[C4 + ch];
    #pragma unroll
    for (int i = 0; i < 8; ++i) {
      int M = lo ? i : (8 + i);
      float v = acc[ct][i] * sc + bi;
      buf[wave][M][ch] = v > 0.f ? v : NEG * v;
    }
  }
  __syncthreads();

  v8f acc2[4];
  #pragma unroll
  for (int ct = 0; ct < 4; ++ct) acc2[ct] = v8zero();
  gemm64(acc2, buf[wave], W3, m, lo);

  #pragma unroll
  for (int ct = 0; ct < 4; ++ct) {
    float s = 0.f, ss = 0.f;
    #pragma unroll
    for (int i = 0; i < 8; ++i) { float v = acc2[ct][i]; s += v; ss += v * v; }
    s  += __shfl_xor(s, 16, 32);
    ss += __shfl_xor(ss, 16, 32);
    if (lo) {
      int ch = ct * 16 + m;
      atomicAdd(&stats[ch], s);
      atomicAdd(&stats[C4 + ch], ss);
    }
  }
}

// K11: full h2 -> bn/lrelu -> h3 -> bn/lrelu -> max over k=32 -> out.
// One point per wave, two 16-row halves.
__global__ void k_out(const float* __restrict__ f1,
                      const int* __restrict__ idx_fps,
                      const float* __restrict__ W2,
                      const float* __restrict__ W3,
                      const float* __restrict__ sb2,
                      const float* __restrict__ sb3,
                      float* __restrict__ out) {
  __shared__ float buf[8][16][C4];
  const int wave = threadIdx.x >> 5;
  const int lane = threadIdx.x & 31;
  const int m = lane & 15;
  const bool lo = lane < 16;
  const int g = blockIdx.x * 8 + wave;
  const int b = g / Nn;

  float mx[4] = {-FLTMAX, -FLTMAX, -FLTMAX, -FLTMAX};
  for (int half = 0; half < 2; ++half) {
    __syncthreads();
    stage_e2(buf[wave], f1, idx_fps, g, b, half * 16 + (lane & 15), lane);
    __syncthreads();

    v8f acc[4];
    #pragma unroll
    for (int ct = 0; ct < 4; ++ct) acc[ct] = v8zero();
    gemm64(acc, buf[wave], W2, m, lo);
    __syncthreads();
    #pragma unroll
    for (int ct = 0; ct < 4; ++ct) {
      int ch = ct * 16 + m;
      float sc = sb2[ch], bi = sb2[C4 + ch];
      #pragma unroll
      for (int i = 0; i < 8; ++i) {
        int M = lo ? i : (8 + i);
        float v = acc[ct][i] * sc + bi;
        buf[wave][M][ch] = v > 0.f ? v : NEG * v;
      }
    }
    __syncthreads();

    v8f acc2[4];
    #pragma unroll
    for (int ct = 0; ct < 4; ++ct) acc2[ct] = v8zero();
    gemm64(acc2, buf[wave], W3, m, lo);

    #pragma unroll
    for (int ct = 0; ct < 4; ++ct) {
      int ch = ct * 16 + m;
      float sc = sb3[ch], bi = sb3[C4 + ch];
      #pragma unroll
      for (int i = 0; i < 8; ++i) {
        float v = acc2[ct][i] * sc + bi;
        v = v > 0.f ? v : NEG * v;
        mx[ct] = fmaxf(mx[ct], v);
      }
    }
  }
  #pragma unroll
  for (int ct = 0; ct < 4; ++ct) {
    float v = fmaxf(mx[ct], __shfl_xor(mx[ct], 16, 32));
    if (lo) out[(size_t)g * OUTd + ct * 16 + m] = v;
  }
}

// ---------------------------------------------------------------------------
extern "C" void kernel_launch(void* const* d_in, const int* in_sizes, int n_in,
                              void* d_out, int out_size, void* d_ws, size_t ws_size,
                              hipStream_t stream) {
  (void)in_sizes; (void)n_in; (void)out_size; (void)ws_size;
  const float* x   = (const float*)d_in[0];
  const float* pos = (const float*)d_in[1];
  const float* W1  = (const float*)d_in[2];
  const float* g1  = (const float*)d_in[3];
  const float* be1 = (const float*)d_in[4];
  const float* W2  = (const float*)d_in[5];
  const float* g2  = (const float*)d_in[6];
  const float* be2 = (const float*)d_in[7];
  const float* W3  = (const float*)d_in[8];
  const float* g3  = (const float*)d_in[9];
  const float* be3 = (const float*)d_in[10];

  float* out     = (float*)d_out;
  float* out_idx = out + (size_t)BN * OUTd;      // idx1 as float, concatenated

  char* ws = (char*)d_ws;
  int*   idx_l  = (int*)ws;                                        // BN*128
  int*   idx1   = (int*)(ws + (size_t)BN * ALLK * 4);              // BN*14
  int*   idxfps = (int*)((char*)idx1 + (size_t)BN * KE * 4);       // BN*32
  float* f1     = (float*)((char*)idxfps + (size_t)BN * Kk * 4);   // BN*32
  float* stats  = (float*)((char*)f1 + (size_t)BN * C2 * 4);       // 320 f
  float* sb     = stats + 320;                                     // 320 f
  float* sum1 = stats;        // sq1 = stats+32
  float* sum2 = stats + 64;   // sq2 = stats+128
  float* sum3 = stats + 192;  // sq3 = stats+256
  float* sb1 = sb;            // scale[32], bias[32]
  float* sb2 = sb + 64;       // scale[64], bias[64]
  float* sb3 = sb + 192;      // scale[64], bias[64]

  hipMemsetAsync(stats, 0, 320 * sizeof(float), stream);

  // 256KB LDS per workgroup: legal on CDNA5 (320KB/WGP); opt in every call.
  const int distLds = 16 * Nn * (int)sizeof(float);
  hipFuncSetAttribute(reinterpret_cast<const void*>(k_dist_pos),
                      hipFuncAttributeMaxDynamicSharedMemorySize, distLds);
  hipFuncSetAttribute(reinterpret_cast<const void*>(k_dist_feat),
                      hipFuncAttributeMaxDynamicSharedMemorySize, distLds);

  k_dist_pos <<<Bb * (Nn / 16), 512, distLds, stream>>>(pos, idx_l, idx1, out_idx);
  k_dist_feat<<<Bb * (Nn / 16), 512, distLds, stream>>>(x, idx1, out_idx);
  k_fps      <<<BN / 8, 256, 0, stream>>>(pos, idx_l, idxfps);

  k_h1_stats <<<(BN * KE / 16) / 8, 256, 0, stream>>>(x, idx1, W1, sum1);
  k_finalize <<<1, 32, 0, stream>>>(sum1, sum1 + 32, g1, be1,
                                    1.f / (float)(BN * KE), C2, sb1);
  k_h1_apply <<<BN / 8, 256, 0, stream>>>(x, idx1, W1, sb1, f1);

  k_h2_stats <<<(BN * Kk / 16) / 8, 256, 0, stream>>>(f1, idxfps, W2, sum2);
  k_finalize <<<1, 64, 0, stream>>>(sum2, sum2 + 64, g2, be2,
                                    1.f / (float)(BN * Kk), C4, sb2);

  k_h3_stats <<<(BN * Kk / 16) / 8, 256, 0, stream>>>(f1, idxfps, W2, W3, sb2, sum3);
  k_finalize <<<1, 64, 0, stream>>>(sum3, sum3 + 64, g3, be3,
                                    1.f / (float)(BN * Kk), C4, sb3);

  k_out      <<<BN / 8, 256, 0, stream>>>(f1, idxfps, W2, W3, sb2, sb3, out);
}